// HardHeatMap_31808527794313
// MI455X (gfx1250) — compile-verified
//
#include <hip/hip_runtime.h>
#include <hip/hip_bf16.h>

#define WDIM 4096
#define HDIM 4096
#define TILE 128
#define TPB  256

typedef int v4i __attribute__((ext_vector_type(4)));
typedef __attribute__((address_space(1))) v4i* global_v4i_ptr;
typedef __attribute__((address_space(3))) v4i* lds_v4i_ptr;

__global__ __launch_bounds__(TPB) void heatmap_tile_kernel(
    const float2* __restrict__ boxes, int n, float* __restrict__ out)
{
    __shared__ float tile[TILE * TILE];  // 64 KB

    const int tid = threadIdx.x;
    const int tx0 = blockIdx.x * TILE;   // image row (W/x dim)
    const int ty0 = blockIdx.y * TILE;   // image col (H/y dim)

    // ---- zero the LDS tile (packed float4 stores) ----
    float4* t4 = (float4*)tile;
    constexpr int ZITERS = (TILE * TILE / 4) / TPB;  // 16
#pragma unroll
    for (int i = 0; i < ZITERS; ++i) {
        t4[tid + i * TPB] = make_float4(0.f, 0.f, 0.f, 0.f);
    }
    __syncthreads();

    // ---- scatter dilated 3x3 footprints into the tile ----
    for (int i = tid; i < n; i += TPB) {
        float2 p = boxes[i];
        int cx = (int)(p.x * (float)WDIM);
        int cy = (int)(p.y * (float)HDIM);
        // match JAX: out-of-bounds scatter updates are dropped
        // (p.x just below 1.0 can round p.x*4096 up to exactly 4096.0f)
        if (cx >= WDIM || cy >= HDIM) continue;
        int rx = cx - tx0;
        int ry = cy - ty0;
        if (rx >= -1 && rx <= TILE && ry >= -1 && ry <= TILE) {
#pragma unroll
            for (int dx = -1; dx <= 1; ++dx) {
                int x = rx + dx;
                if (x < 0 || x >= TILE) continue;
#pragma unroll
                for (int dy = -1; dy <= 1; ++dy) {
                    int y = ry + dy;
                    if (y >= 0 && y < TILE) {
                        tile[x * TILE + y] = 1.0f;  // benign race: same value
                    }
                }
            }
        }
    }
    __syncthreads();

    // ---- stream the tile to global memory ----
    constexpr int CHUNKS = TILE * TILE * 4 / 16;  // 4096 x 16B chunks
    constexpr int SITERS = CHUNKS / TPB;          // 16 per thread

#if __has_builtin(__builtin_amdgcn_global_store_async_from_lds_b128)
    // CDNA5 async LDS->global DMA path (tracked by ASYNCcnt)
#pragma unroll
    for (int i = 0; i < SITERS; ++i) {
        int c   = tid + i * TPB;
        int row = c >> 5;              // 32 chunks of 16B per 128-col row
        int col = (c & 31) << 2;       // float column within row
        float* g = out + (size_t)(tx0 + row) * (size_t)HDIM + (size_t)(ty0 + col);
        __builtin_amdgcn_global_store_async_from_lds_b128(
            (global_v4i_ptr)g, (lds_v4i_ptr)&tile[c * 4], /*offset=*/0, /*cpol=*/0);
    }
#if __has_builtin(__builtin_amdgcn_s_wait_asynccnt)
    __builtin_amdgcn_s_wait_asynccnt(0);
#endif
#else
    // Fallback: LDS -> VGPR -> global, 128-bit each way
#pragma unroll
    for (int i = 0; i < SITERS; ++i) {
        int c   = tid + i * TPB;
        int row = c >> 5;
        int col = (c & 31) << 2;
        float4 v = *(const float4*)&tile[c * 4];
        *(float4*)(out + (size_t)(tx0 + row) * (size_t)HDIM + (size_t)(ty0 + col)) = v;
    }
#endif
}

extern "C" void kernel_launch(void* const* d_in, const int* in_sizes, int n_in,
                              void* d_out, int out_size, void* d_ws, size_t ws_size,
                              hipStream_t stream) {
    const float2* boxes = (const float2*)d_in[0];
    const int n = in_sizes[0] / 2;  // 20000 (x,y) pairs
    float* out = (float*)d_out;    // (1,1,4096,4096) fp32, row-major

    dim3 grid(WDIM / TILE, HDIM / TILE);  // 32 x 32 tiles
    heatmap_tile_kernel<<<grid, TPB, 0, stream>>>(boxes, n, out);
}